// AngularAggLayer_68247030333958
// MI455X (gfx1250) — compile-verified
//
#include <hip/hip_runtime.h>
#include <math.h>

#define NUM_CLASS 16
#define DFEAT     128
#define EPSV      1e-5f
#define CAPNZ     2048   // max nonzeros per A-row buffered in LDS (p=0.01 -> mean 61)

typedef float v2f __attribute__((ext_vector_type(2)));
typedef float v8f __attribute__((ext_vector_type(8)));

// ---------------------------------------------------------------------------
// K1: complex normalize x -> nf, and s_feat[n] = <nf[n,:], p1> (complex dot)
// one block (128 threads) per row
// ---------------------------------------------------------------------------
__global__ void k_normalize(const float* __restrict__ xr, const float* __restrict__ xi,
                            const float* __restrict__ pr, const float* __restrict__ pi,
                            float* __restrict__ nfr, float* __restrict__ nfi,
                            float* __restrict__ sfr, float* __restrict__ sfi) {
    __shared__ float rre[DFEAT], rim[DFEAT];
    const int n = blockIdx.x, t = threadIdx.x;
    const size_t idx = (size_t)n * DFEAT + t;
    float a = xr[idx], b = xi[idx];
    float inv = 1.0f / (sqrtf(a * a + b * b) + EPSV);
    float nr = a * inv, ni = b * inv;
    nfr[idx] = nr;
    nfi[idx] = ni;
    float p1r = pr[t], p1i = pi[t];
    rre[t] = nr * p1r - ni * p1i;
    rim[t] = nr * p1i + ni * p1r;
    __syncthreads();
    for (int s = DFEAT / 2; s > 0; s >>= 1) {
        if (t < s) { rre[t] += rre[t + s]; rim[t] += rim[t + s]; }
        __syncthreads();
    }
    if (t == 0) { sfr[n] = rre[0]; sfi[n] = rim[0]; }
}

// ---------------------------------------------------------------------------
// K2: class sums via V_WMMA_F32_16X16X4_F32.
// sum_by_label = one_hot(labels)^T (16 x N) @ nf (N x 128), per real/imag part.
// 16 tiles of 16x16 (8 feature blocks x 2 parts); one wave32 per tile.
// A-frag (16x4 f32): lanes 0-15 hold M=0..15; VGPR v holds K = 2*hi + v.
// B-frag (4x16 f32): lane cols, mirrored K layout.
// ---------------------------------------------------------------------------
__global__ void k_classsum_wmma(const float* __restrict__ nfr, const float* __restrict__ nfi,
                                const int* __restrict__ labels, int N,
                                float* __restrict__ csr, float* __restrict__ csi) {
    const int part = blockIdx.x & 1;       // 0: real, 1: imag
    const int fb   = blockIdx.x >> 1;      // feature block, 16 cols each
    const float* __restrict__ src = part ? nfi : nfr;
    float* __restrict__ dst = part ? csi : csr;

    const int l   = threadIdx.x;           // 0..31 (one wave)
    const int m   = l & 15;                // class row this lane holds in A
    const int hi  = l >> 4;                // half-wave selector
    const int col = fb * 16 + (l & 15);    // feature column this lane holds in B/D

    v8f acc = {};
    for (int k0 = 0; k0 < N; k0 += 4) {
        const int ka  = k0 + 2 * hi;
        const int la0 = labels[ka];
        const int la1 = labels[ka + 1];
        v2f afrag, bfrag;
        afrag[0] = (la0 == m) ? 1.0f : 0.0f;
        afrag[1] = (la1 == m) ? 1.0f : 0.0f;
        bfrag[0] = src[(size_t)ka * DFEAT + col];
        bfrag[1] = src[(size_t)(ka + 1) * DFEAT + col];
        acc = __builtin_amdgcn_wmma_f32_16x16x4_f32(
            /*neg_a=*/false, afrag, /*neg_b=*/false, bfrag,
            /*c_mod=*/(short)0, acc, /*reuse_a=*/false, /*reuse_b=*/false);
    }
    // C/D layout: VGPR v -> row M = v + 8*hi, lane col = l&15
#pragma unroll
    for (int v = 0; v < 8; ++v) {
        dst[(v + 8 * hi) * DFEAT + col] = acc[v];
    }
}

// ---------------------------------------------------------------------------
// K3 (single block, 256 thr): counts, column sums S = sum_c class_sum[c],
// M matrix (faithful: M[i,j] = -theta[j(j-1)/2+i] for i<j, else 0),
// s_cent[c] = <class_sum[c]/count[c], p2> (complex dot)
// ---------------------------------------------------------------------------
__global__ void k_stats(const float* __restrict__ csr, const float* __restrict__ csi,
                        const int* __restrict__ labels, int N,
                        const float* __restrict__ theta,
                        const float* __restrict__ pr, const float* __restrict__ pi,
                        float* __restrict__ colr, float* __restrict__ coli,
                        float* __restrict__ scr, float* __restrict__ sci,
                        float* __restrict__ Mout) {
    __shared__ int cnt[NUM_CLASS];
    const int t = threadIdx.x;
    if (t < NUM_CLASS) cnt[t] = 0;
    __syncthreads();
    for (int n = t; n < N; n += blockDim.x) atomicAdd(&cnt[labels[n]], 1);
    __syncthreads();

    // column sums of nf = sum over classes of class sums
    {
        const int part = t >> 7, d = t & 127;
        const float* cs = part ? csi : csr;
        float s = 0.f;
        for (int c = 0; c < NUM_CLASS; ++c) s += cs[c * DFEAT + d];
        if (part) coli[d] = s; else colr[d] = s;
    }
    // M: the reference's second .set overwrites the upper triangle with
    // -theta in column-major triu order; lower triangle stays zero.
    {
        const int i = t >> 4, j = t & 15;
        float v = 0.f;
        if (i < j) v = -theta[(j * (j - 1)) / 2 + i];
        Mout[i * NUM_CLASS + j] = v;
    }
    // s_cent
    if (t < NUM_CLASS) {
        const int c = t;
        const float invc = 1.0f / (float)cnt[c];
        float sr = 0.f, si = 0.f;
        for (int d = 0; d < DFEAT; ++d) {
            float mr = csr[c * DFEAT + d] * invc;
            float mi = csi[c * DFEAT + d] * invc;
            float p2r = pr[DFEAT + d], p2i = pi[DFEAT + d];
            sr += mr * p2r - mi * p2i;
            si += mr * p2i + mi * p2r;
        }
        scr[c] = sr;
        sci[c] = si;
    }
}

// ---------------------------------------------------------------------------
// K4: fake_label[n] = argmax_c |s_feat[n] + s_cent[c]| (squared-mag compare)
// ---------------------------------------------------------------------------
__global__ void k_fakelabel(const float* __restrict__ sfr, const float* __restrict__ sfi,
                            const float* __restrict__ scr, const float* __restrict__ sci,
                            int N, int* __restrict__ fl) {
    const int n = blockIdx.x * blockDim.x + threadIdx.x;
    if (n >= N) return;
    const float sr = sfr[n], si = sfi[n];
    float best = -1.f;
    int arg = 0;
#pragma unroll
    for (int c = 0; c < NUM_CLASS; ++c) {
        float tr = sr + scr[c], ti = si + sci[c];
        float v = tr * tr + ti * ti;
        if (v > best) { best = v; arg = c; }   // strict > == first-max (argmax)
    }
    fl[n] = arg;
}

// ---------------------------------------------------------------------------
// K5 (hot loop, memory-bound on A: 151 MB @ 23.3 TB/s):
//   message[i] = colsum + sum_{A[i,j]>0} (e^{i*M[fl_i,fl_j]} - 1) * nf[j]
//   out[i]     = message[i] / (|message[i]| + eps)
// Phase 1: 128 threads stream A-row as float4 (+prefetch), compact nnz to LDS.
// Phase 2: thread t owns feature dim t; coalesced gathers of nf (L2-resident).
// ---------------------------------------------------------------------------
__global__ void k_aggregate(const float* __restrict__ A,
                            const float* __restrict__ nfr, const float* __restrict__ nfi,
                            const float* __restrict__ colr, const float* __restrict__ coli,
                            const int* __restrict__ fl, const float* __restrict__ M,
                            int N, float* __restrict__ out) {
    __shared__ float wre[CAPNZ], wim[CAPNZ];
    __shared__ int   widx[CAPNZ];
    __shared__ int   nnz;
    __shared__ float Mrow[NUM_CLASS];

    const int i = blockIdx.x, t = threadIdx.x;   // 128 threads
    if (t == 0) nnz = 0;
    if (t < NUM_CLASS) Mrow[t] = M[fl[i] * NUM_CLASS + t];
    __syncthreads();

    const float4* __restrict__ Arow = (const float4*)(A + (size_t)i * N);
    const int nq = N >> 2;
    for (int q = t; q < nq; q += blockDim.x) {
        if (q + (int)blockDim.x < nq)
            __builtin_prefetch((const void*)&Arow[q + blockDim.x], 0, 0);
        float4 a4 = Arow[q];
        float av[4] = { a4.x, a4.y, a4.z, a4.w };
#pragma unroll
        for (int c = 0; c < 4; ++c) {
            if (av[c] > 0.f) {
                int j = q * 4 + c;
                float th = Mrow[fl[j]];
                int slot = atomicAdd(&nnz, 1);
                if (slot < CAPNZ) {
                    wre[slot]  = cosf(th) - 1.0f;
                    wim[slot]  = sinf(th);
                    widx[slot] = j;
                }
            }
        }
    }
    __syncthreads();

    int m = nnz;
    if (m > CAPNZ) m = CAPNZ;
    float ar = colr[t], ai = coli[t];
    for (int k = 0; k < m; ++k) {
        const float wr = wre[k], wi = wim[k];
        const int j = widx[k];
        const float nr = nfr[(size_t)j * DFEAT + t];
        const float ni = nfi[(size_t)j * DFEAT + t];
        ar += wr * nr - wi * ni;
        ai += wr * ni + wi * nr;
    }
    const float inv = 1.0f / (sqrtf(ar * ar + ai * ai) + EPSV);
    const size_t o = ((size_t)i * DFEAT + t) * 2;   // complex64 interleaved
    out[o]     = ar * inv;
    out[o + 1] = ai * inv;
}

// ---------------------------------------------------------------------------
extern "C" void kernel_launch(void* const* d_in, const int* in_sizes, int n_in,
                              void* d_out, int out_size, void* d_ws, size_t ws_size,
                              hipStream_t stream) {
    const float* xr    = (const float*)d_in[0];
    const float* xi    = (const float*)d_in[1];
    const float* A     = (const float*)d_in[2];
    const float* theta = (const float*)d_in[3];
    const float* pr    = (const float*)d_in[4];
    const float* pi    = (const float*)d_in[5];
    const int*   lab   = (const int*)d_in[6];
    const int N = in_sizes[0] / DFEAT;

    float* ws = (float*)d_ws;
    const size_t NF = (size_t)N * DFEAT;
    float* nfr = ws;  ws += NF;
    float* nfi = ws;  ws += NF;
    float* csr = ws;  ws += NUM_CLASS * DFEAT;
    float* csi = ws;  ws += NUM_CLASS * DFEAT;
    float* colr = ws; ws += DFEAT;
    float* coli = ws; ws += DFEAT;
    float* sfr = ws;  ws += N;
    float* sfi = ws;  ws += N;
    float* scr = ws;  ws += NUM_CLASS;
    float* sci = ws;  ws += NUM_CLASS;
    float* M   = ws;  ws += NUM_CLASS * NUM_CLASS;
    int*   fl  = (int*)ws;

    k_normalize<<<N, DFEAT, 0, stream>>>(xr, xi, pr, pi, nfr, nfi, sfr, sfi);
    k_classsum_wmma<<<16, 32, 0, stream>>>(nfr, nfi, lab, N, csr, csi);
    k_stats<<<1, 256, 0, stream>>>(csr, csi, lab, N, theta, pr, pi,
                                   colr, coli, scr, sci, M);
    k_fakelabel<<<(N + 255) / 256, 256, 0, stream>>>(sfr, sfi, scr, sci, N, fl);
    k_aggregate<<<N, DFEAT, 0, stream>>>(A, nfr, nfi, colr, coli, fl, M, N,
                                         (float*)d_out);
}